// V2GLayer_17669495456075
// MI455X (gfx1250) — compile-verified
//
#include <hip/hip_runtime.h>

#define DV 128
#define DG 256
#define G  4096
#define K3 384          // 3*DV
#define NPW 128         // nodes per wave in the reduction pass

typedef float v2f __attribute__((ext_vector_type(2)));
typedef float v4f __attribute__((ext_vector_type(4)));
typedef float v8f __attribute__((ext_vector_type(8)));

// Order-preserving float <-> uint key (so min/max can use native u32 atomics)
__device__ __forceinline__ unsigned fkey(float f) {
    unsigned u = __float_as_uint(f);
    return (u & 0x80000000u) ? ~u : (u | 0x80000000u);
}
__device__ __forceinline__ float kinv(unsigned k) {
    unsigned u = (k & 0x80000000u) ? (k ^ 0x80000000u) : ~k;
    return __uint_as_float(u);
}

// ---------------- init workspace ----------------
__global__ __launch_bounds__(256) void init_ws(float* __restrict__ sum,
                                               unsigned* __restrict__ mink,
                                               unsigned* __restrict__ maxk,
                                               unsigned* __restrict__ cnt) {
    int idx = blockIdx.x * 256 + threadIdx.x;
    if (idx < G * DV) {
        sum[idx]  = 0.0f;
        mink[idx] = 0xFFFFFFFFu;   // identity for key-min
        maxk[idx] = 0u;            // identity for key-max
    }
    if (idx < G) cnt[idx] = 0u;
}

// ---------------- streaming segment reduce ----------------
__device__ __forceinline__ void flush_seg(int g, int lane, const v4f& s,
                                          const v4f& mn, const v4f& mx, unsigned c,
                                          float* __restrict__ sum, unsigned* __restrict__ mink,
                                          unsigned* __restrict__ maxk, unsigned* __restrict__ cnt) {
    size_t o = (size_t)g * DV + (size_t)lane * 4;
    atomicAdd(&sum[o + 0], s.x);  atomicAdd(&sum[o + 1], s.y);
    atomicAdd(&sum[o + 2], s.z);  atomicAdd(&sum[o + 3], s.w);
    atomicMin(&mink[o + 0], fkey(mn.x)); atomicMin(&mink[o + 1], fkey(mn.y));
    atomicMin(&mink[o + 2], fkey(mn.z)); atomicMin(&mink[o + 3], fkey(mn.w));
    atomicMax(&maxk[o + 0], fkey(mx.x)); atomicMax(&maxk[o + 1], fkey(mx.y));
    atomicMax(&maxk[o + 2], fkey(mx.z)); atomicMax(&maxk[o + 3], fkey(mx.w));
    if (lane == 0) atomicAdd(&cnt[g], c);
}

__global__ __launch_bounds__(256) void seg_reduce(const v4f* __restrict__ fv4,
                                                  const int* __restrict__ seg,
                                                  float* __restrict__ sum,
                                                  unsigned* __restrict__ mink,
                                                  unsigned* __restrict__ maxk,
                                                  unsigned* __restrict__ cnt, int n) {
    const int  lane = threadIdx.x & 31;
    const long wave = (long)blockIdx.x * (blockDim.x >> 5) + (threadIdx.x >> 5);
    long base = wave * NPW;
    if (base >= n) return;

    const float INF = __uint_as_float(0x7F800000u);
    v4f s  = {0.f, 0.f, 0.f, 0.f};
    v4f mn = {INF, INF, INF, INF};
    v4f mx = {-INF, -INF, -INF, -INF};
    unsigned c = 0;
    int cur = seg[base];

    for (int i = 0; i < NPW; i += 32) {
        long nb = base + i;
        if (nb >= n) break;
        // coalesced, non-temporal segment-id fetch, broadcast via shuffle
        int sgl = (nb + lane < n) ? __builtin_nontemporal_load(seg + nb + lane) : 0;
        int lim = (n - nb < 32) ? (int)(n - nb) : 32;
        for (int j = 0; j < lim; ++j) {
            int g = __shfl(sgl, j, 32);
            if (g != cur) {
                flush_seg(cur, lane, s, mn, mx, c, sum, mink, maxk, cnt);
                s  = (v4f){0.f, 0.f, 0.f, 0.f};
                mn = (v4f){INF, INF, INF, INF};
                mx = (v4f){-INF, -INF, -INF, -INF};
                c = 0;
                cur = g;
            }
            // single-use 256MB stream: non-temporal b128 load keeps L2 for the
            // atomic reduction targets and the GEMM operands
            v4f v = __builtin_nontemporal_load(fv4 + (size_t)(nb + j) * (DV / 4) + lane);
            s.x += v.x; s.y += v.y; s.z += v.z; s.w += v.w;
            mn.x = fminf(mn.x, v.x); mn.y = fminf(mn.y, v.y);
            mn.z = fminf(mn.z, v.z); mn.w = fminf(mn.w, v.w);
            mx.x = fmaxf(mx.x, v.x); mx.y = fmaxf(mx.y, v.y);
            mx.z = fmaxf(mx.z, v.z); mx.w = fmaxf(mx.w, v.w);
            ++c;
        }
    }
    if (c) flush_seg(cur, lane, s, mn, mx, c, sum, mink, maxk, cnt);
}

// ---------------- finalize z = [mean | min | max] ----------------
__global__ __launch_bounds__(256) void finalize_z(const float* __restrict__ sum,
                                                  const unsigned* __restrict__ mink,
                                                  const unsigned* __restrict__ maxk,
                                                  const unsigned* __restrict__ cnt,
                                                  float* __restrict__ z) {
    int idx = blockIdx.x * 256 + threadIdx.x;
    if (idx >= G * DV) return;
    int g = idx >> 7, c = idx & (DV - 1);
    unsigned cc = cnt[g];
    float denom = (float)(cc > 1u ? cc : 1u);
    size_t zb = (size_t)g * K3;
    z[zb + c]          = sum[idx] / denom;
    z[zb + DV + c]     = kinv(mink[idx]);
    z[zb + 2 * DV + c] = kinv(maxk[idx]);
}

// ---------------- GEMM: out[g][o] = z[g][:] . W[o][:] + b[o] via fp32 WMMA ----------------
// One wave computes a 16(g) x 64(o) patch: 4 accumulator tiles, A reused across them.
// A layout (16x4 f32): lane = (half,r); a.{x,y} = A[r][2*half + {0,1}]
// B layout (4x16 f32): b.{x,y} = B[2*half + {0,1}][r] = W[o0+r][k0+2*half+{0,1}]
// C/D layout: VGPR j -> row M = j + 8*half, col N = r
__global__ __launch_bounds__(256) void gemm_wmma_f32(const float* __restrict__ z,
                                                     const float* __restrict__ W,
                                                     const float* __restrict__ bias,
                                                     float* __restrict__ out) {
    const int lane = threadIdx.x & 31;
    const int wid  = (blockIdx.x * 256 + threadIdx.x) >> 5;   // 0..1023
    const int gTile = wid >> 2;                                // 0..255
    const int oGrp  = wid & 3;                                 // 0..3
    const int half  = lane >> 4;
    const int r     = lane & 15;
    const int g0 = gTile * 16;
    const int o0 = oGrp * 64;

    v8f zero = {};
    v8f acc[4];
#pragma unroll
    for (int t = 0; t < 4; ++t) acc[t] = zero;

    const v2f* za  = (const v2f*)(z + (size_t)(g0 + r) * K3 + 2 * half);
    const v2f* wb0 = (const v2f*)(W + (size_t)(o0 + 0 * 16 + r) * K3 + 2 * half);
    const v2f* wb1 = (const v2f*)(W + (size_t)(o0 + 1 * 16 + r) * K3 + 2 * half);
    const v2f* wb2 = (const v2f*)(W + (size_t)(o0 + 2 * 16 + r) * K3 + 2 * half);
    const v2f* wb3 = (const v2f*)(W + (size_t)(o0 + 3 * 16 + r) * K3 + 2 * half);

    for (int k0 = 0; k0 < K3; k0 += 4) {
        int ki = k0 >> 1;
        v2f a  = za[ki];
        v2f b0 = wb0[ki], b1 = wb1[ki], b2 = wb2[ki], b3 = wb3[ki];
        acc[0] = __builtin_amdgcn_wmma_f32_16x16x4_f32(false, a, false, b0, (short)0, acc[0], false, false);
        acc[1] = __builtin_amdgcn_wmma_f32_16x16x4_f32(false, a, false, b1, (short)0, acc[1], false, false);
        acc[2] = __builtin_amdgcn_wmma_f32_16x16x4_f32(false, a, false, b2, (short)0, acc[2], false, false);
        acc[3] = __builtin_amdgcn_wmma_f32_16x16x4_f32(false, a, false, b3, (short)0, acc[3], false, false);
    }

#pragma unroll
    for (int t = 0; t < 4; ++t) {
        int ocol = o0 + t * 16 + r;
        float bv = bias[ocol];
#pragma unroll
        for (int j = 0; j < 8; ++j) {
            int grow = g0 + j + 8 * half;
            out[(size_t)grow * DG + ocol] = acc[t][j] + bv;
        }
    }
}

// ---------------- launch ----------------
extern "C" void kernel_launch(void* const* d_in, const int* in_sizes, int n_in,
                              void* d_out, int out_size, void* d_ws, size_t ws_size,
                              hipStream_t stream) {
    const float* fv  = (const float*)d_in[0];
    const int*   seg = (const int*)d_in[1];
    // d_in[2] = num_graphs (compile-time constant G here)
    const float* W   = (const float*)d_in[3];
    const float* b   = (const float*)d_in[4];
    float* out = (float*)d_out;
    int n = in_sizes[1];   // number of nodes

    char* p = (char*)d_ws;
    float*    sum  = (float*)p;    p += (size_t)G * DV * sizeof(float);
    unsigned* mink = (unsigned*)p; p += (size_t)G * DV * sizeof(unsigned);
    unsigned* maxk = (unsigned*)p; p += (size_t)G * DV * sizeof(unsigned);
    unsigned* cnt  = (unsigned*)p; p += (size_t)G * sizeof(unsigned);
    float*    z    = (float*)p;    // G * K3 floats

    init_ws<<<(G * DV + 255) / 256, 256, 0, stream>>>(sum, mink, maxk, cnt);

    int waves  = (n + NPW - 1) / NPW;
    int blocks = (waves + 7) / 8;          // 8 waves (256 threads) per block
    seg_reduce<<<blocks, 256, 0, stream>>>((const v4f*)fv, seg, sum, mink, maxk, cnt, n);

    finalize_z<<<(G * DV + 255) / 256, 256, 0, stream>>>(sum, mink, maxk, cnt, z);

    // 1024 waves total: 256 g-tiles x 4 o-groups; 8 waves per block -> 128 blocks
    gemm_wmma_f32<<<(G / 16) * (DG / 64) / 8, 256, 0, stream>>>(z, W, b, out);
}